// Qwen3OmniMoeSparseMoeBlock_26663156973648
// MI455X (gfx1250) — compile-verified
//
#include <hip/hip_runtime.h>
#include <math.h>

typedef float v2f __attribute__((ext_vector_type(2)));
typedef float v8f __attribute__((ext_vector_type(8)));

#define T_TOK 4096
#define HDIM  2048
#define NEXP  8
#define IDIM  768
#define I2DIM 1536
#define MAXENT 8704          // 2*T + 8*64 padding
#define MAX_TT 64            // max 64-token tiles per expert (4096/64)
#define LDSPAD 36            // row stride in floats: 16B-aligned, conflict-free

// ---------------- workspace layout (int units) ----------------
// [0..7]   cursor (router counts)  [8..15] cnt  [16..23] cntpad  [24..31] offs
// [32..39] cursor2
// [40..]   pair_e[2T], list_tok[MAXENT], pair_w[2T], list_w[MAXENT], inter[MAXENT*IDIM]

__global__ __launch_bounds__(256)
void init_kernel(float* outp, int* cursor) {
    size_t i = (size_t)blockIdx.x * blockDim.x + threadIdx.x;
    if (i < 8) cursor[i] = 0;
    const size_t n = (size_t)T_TOK * HDIM;
    for (size_t k = i; k < n; k += (size_t)gridDim.x * blockDim.x) outp[k] = 0.0f;
}

__global__ __launch_bounds__(256)
void router_kernel(const float* __restrict__ x, const float* __restrict__ gw,
                   float* __restrict__ out_logits,
                   int* __restrict__ cursor, int* __restrict__ pair_e,
                   float* __restrict__ pair_w) {
    __shared__ float red[NEXP][256];
    const int t = blockIdx.x;
    const int tid = threadIdx.x;
    float acc[NEXP];
#pragma unroll
    for (int e = 0; e < NEXP; ++e) acc[e] = 0.0f;
#pragma unroll
    for (int j = 0; j < 8; ++j) {
        int h = tid + j * 256;
        float xv = x[(size_t)t * HDIM + h];
        const float* g = gw + (size_t)h * NEXP;
#pragma unroll
        for (int e = 0; e < NEXP; ++e) acc[e] += xv * g[e];
    }
#pragma unroll
    for (int e = 0; e < NEXP; ++e) red[e][tid] = acc[e];
    __syncthreads();
    for (int s = 128; s > 0; s >>= 1) {
        if (tid < s) {
#pragma unroll
            for (int e = 0; e < NEXP; ++e) red[e][tid] += red[e][tid + s];
        }
        __syncthreads();
    }
    if (tid == 0) {
        float lg[NEXP];
#pragma unroll
        for (int e = 0; e < NEXP; ++e) { lg[e] = red[e][0]; out_logits[t * NEXP + e] = lg[e]; }
        float m = lg[0];
#pragma unroll
        for (int e = 1; e < NEXP; ++e) m = fmaxf(m, lg[e]);
        float p[NEXP], s = 0.0f;
#pragma unroll
        for (int e = 0; e < NEXP; ++e) { p[e] = expf(lg[e] - m); s += p[e]; }
#pragma unroll
        for (int e = 0; e < NEXP; ++e) p[e] /= s;
        int m0 = 0;
#pragma unroll
        for (int e = 1; e < NEXP; ++e) if (p[e] > p[m0]) m0 = e;
        int m1 = (m0 == 0) ? 1 : 0;
#pragma unroll
        for (int e = 0; e < NEXP; ++e) if (e != m0 && p[e] > p[m1]) m1 = e;
        float w0 = p[m0], w1 = p[m1], ss = w0 + w1;
        w0 /= ss; w1 /= ss;
        pair_e[2 * t] = m0; pair_e[2 * t + 1] = m1;
        pair_w[2 * t] = w0; pair_w[2 * t + 1] = w1;
        atomicAdd(&cursor[m0], 1);
        atomicAdd(&cursor[m1], 1);
    }
}

__global__ __launch_bounds__(512)
void scan_pad_kernel(int* ws, int* __restrict__ list_tok, float* __restrict__ list_w) {
    __shared__ int scnt[NEXP], spad[NEXP], soffs[NEXP];
    int* cursor = ws;
    int* cnt = ws + 8;
    int* cntpad = ws + 16;
    int* offs = ws + 24;
    int* cursor2 = ws + 32;
    if (threadIdx.x == 0) {
        int run = 0;
        for (int e = 0; e < NEXP; ++e) {
            int c = cursor[e];
            int p = (c + 63) & ~63;
            cnt[e] = c; cntpad[e] = p; offs[e] = run;
            scnt[e] = c; spad[e] = p; soffs[e] = run;
            cursor2[e] = 0;
            run += p;
        }
    }
    __syncthreads();
    int e = threadIdx.x >> 6;
    int j = threadIdx.x & 63;
    int padn = spad[e] - scnt[e];
    if (j < padn) {
        int idx = soffs[e] + scnt[e] + j;
        list_tok[idx] = 0;
        list_w[idx] = 0.0f;
    }
}

__global__ __launch_bounds__(256)
void scatter_kernel(const int* __restrict__ pair_e, const float* __restrict__ pair_w,
                    const int* __restrict__ offs, int* cursor2,
                    int* __restrict__ list_tok, float* __restrict__ list_w) {
    int t = blockIdx.x * blockDim.x + threadIdx.x;
    if (t >= T_TOK) return;
#pragma unroll
    for (int k = 0; k < 2; ++k) {
        int e = pair_e[2 * t + k];
        int pos = atomicAdd(&cursor2[e], 1);
        int idx = offs[e] + pos;
        list_tok[idx] = t;
        list_w[idx] = pair_w[2 * t + k];
    }
}

// gate_up GEMM + SiLU: 64 tokens x 32 intermediate per block; 256 thr / 8 waves.
// wave w: mt = w&3 (token sub-tile), gu = w>>2 (gate/up); 2 N sub-tiles per wave
// sharing the A fragment across two v_wmma_f32_16x16x4_f32.
__global__ __launch_bounds__(256)
void gemm1_kernel(const float* __restrict__ X, const float* __restrict__ GUP,
                  const int* __restrict__ cntpad, const int* __restrict__ offs,
                  const int* __restrict__ list_tok, float* __restrict__ INTER) {
    const int ex = blockIdx.x >> 6;      // /MAX_TT
    const int tt = blockIdx.x & (MAX_TT - 1);
    const int cpad = cntpad[ex];
    if (tt * 64 >= cpad) return;
    const int i0 = blockIdx.y * 32;
    const int base = offs[ex] + tt * 64;

    __shared__ int   toks[64];
    __shared__ float xs[64][LDSPAD];
    __shared__ float bs[64][LDSPAD];
    __shared__ float stash[2][64][32];

    const int tid  = threadIdx.x;
    const int lane = tid & 31;
    const int wave = tid >> 5;
    const int mt = wave & 3;             // token sub-tile 0..3
    const int gu = wave >> 2;            // 0=gate, 1=up
    const int hi = lane >> 4;
    const int lo = lane & 15;

    if (tid < 64) toks[tid] = list_tok[base + tid];
    __syncthreads();

    // staging: 64 rows x 8 float4-quads; thread -> (row = tid/8 + 32j, quad = tid%8)
    const int c4 = (tid & 7) * 4;
    const float* xrow[2];
    const float* brow[2];
#pragma unroll
    for (int j = 0; j < 2; ++j) {
        int r = (tid >> 3) + 32 * j;
        xrow[j] = X + (size_t)toks[r] * HDIM;
        int f = (r < 32) ? (i0 + r) : (IDIM + i0 + (r - 32));
        brow[j] = GUP + ((size_t)ex * I2DIM + f) * HDIM;
    }

    v8f acc0 = {0.f, 0.f, 0.f, 0.f, 0.f, 0.f, 0.f, 0.f};
    v8f acc1 = acc0;

    for (int kc = 0; kc < HDIM; kc += 32) {
#pragma unroll
        for (int j = 0; j < 2; ++j) {
            int r = (tid >> 3) + 32 * j;
            *(float4*)&xs[r][c4] = *(const float4*)(xrow[j] + kc + c4);
            *(float4*)&bs[r][c4] = *(const float4*)(brow[j] + kc + c4);
        }
        __syncthreads();
#pragma unroll
        for (int s = 0; s < 8; ++s) {
            const int kl = 4 * s + 2 * hi;
            v2f a, b0, b1;
            a.x  = xs[mt * 16 + lo][kl];
            a.y  = xs[mt * 16 + lo][kl + 1];
            b0.x = bs[gu * 32 + lo][kl];
            b0.y = bs[gu * 32 + lo][kl + 1];
            b1.x = bs[gu * 32 + 16 + lo][kl];
            b1.y = bs[gu * 32 + 16 + lo][kl + 1];
            acc0 = __builtin_amdgcn_wmma_f32_16x16x4_f32(
                false, a, false, b0, (short)0, acc0, false, false);
            acc1 = __builtin_amdgcn_wmma_f32_16x16x4_f32(
                false, a, false, b1, (short)0, acc1, false, false);
        }
        __syncthreads();
    }

#pragma unroll
    for (int v = 0; v < 8; ++v) {
        stash[gu][mt * 16 + v + 8 * hi][lo]      = acc0[v];
        stash[gu][mt * 16 + v + 8 * hi][16 + lo] = acc1[v];
    }
    __syncthreads();

#pragma unroll
    for (int j = 0; j < 8; ++j) {
        int e = tid + j * 256;
        int r = e >> 5, c = e & 31;
        float g = stash[0][r][c];
        float u = stash[1][r][c];
        float sg = g / (1.0f + expf(-g));
        INTER[(size_t)(base + r) * IDIM + i0 + c] = sg * u;
    }
}

// down GEMM + weighted atomic scatter: 64 tokens x 64 H per block; 256 thr / 8 waves.
// wave w: mt = w&3, nh = w>>1... nh = w>>2 in {0,1}; 2 N sub-tiles per wave share A.
__global__ __launch_bounds__(256)
void gemm2_kernel(const float* __restrict__ INTER, const float* __restrict__ WD,
                  const int* __restrict__ cntpad, const int* __restrict__ offs,
                  const int* __restrict__ list_tok, const float* __restrict__ list_w,
                  float* __restrict__ OUT) {
    const int ex = blockIdx.x >> 6;
    const int tt = blockIdx.x & (MAX_TT - 1);
    const int cpad = cntpad[ex];
    if (tt * 64 >= cpad) return;
    const int h0 = blockIdx.y * 64;
    const int base = offs[ex] + tt * 64;

    __shared__ int   toks[64];
    __shared__ float wts[64];
    __shared__ float as_[64][LDSPAD];
    __shared__ float bs[64][LDSPAD];

    const int tid  = threadIdx.x;
    const int lane = tid & 31;
    const int wave = tid >> 5;
    const int mt = wave & 3;             // token sub-tile 0..3
    const int nh = wave >> 2;            // h half 0..1
    const int hi = lane >> 4;
    const int lo = lane & 15;

    if (tid < 64) {
        toks[tid] = list_tok[base + tid];
        wts[tid]  = list_w[base + tid];
    }
    __syncthreads();

    const int c4 = (tid & 7) * 4;
    const float* arow[2];
    const float* brow[2];
#pragma unroll
    for (int j = 0; j < 2; ++j) {
        int r = (tid >> 3) + 32 * j;
        arow[j] = INTER + (size_t)(base + r) * IDIM;
        brow[j] = WD + ((size_t)ex * HDIM + h0 + r) * IDIM;
    }

    v8f acc0 = {0.f, 0.f, 0.f, 0.f, 0.f, 0.f, 0.f, 0.f};
    v8f acc1 = acc0;

    for (int kc = 0; kc < IDIM; kc += 32) {
#pragma unroll
        for (int j = 0; j < 2; ++j) {
            int r = (tid >> 3) + 32 * j;
            *(float4*)&as_[r][c4] = *(const float4*)(arow[j] + kc + c4);
            *(float4*)&bs[r][c4]  = *(const float4*)(brow[j] + kc + c4);
        }
        __syncthreads();
#pragma unroll
        for (int s = 0; s < 8; ++s) {
            const int kl = 4 * s + 2 * hi;
            v2f a, b0, b1;
            a.x  = as_[mt * 16 + lo][kl];
            a.y  = as_[mt * 16 + lo][kl + 1];
            b0.x = bs[nh * 32 + lo][kl];
            b0.y = bs[nh * 32 + lo][kl + 1];
            b1.x = bs[nh * 32 + 16 + lo][kl];
            b1.y = bs[nh * 32 + 16 + lo][kl + 1];
            acc0 = __builtin_amdgcn_wmma_f32_16x16x4_f32(
                false, a, false, b0, (short)0, acc0, false, false);
            acc1 = __builtin_amdgcn_wmma_f32_16x16x4_f32(
                false, a, false, b1, (short)0, acc1, false, false);
        }
        __syncthreads();
    }

    const int hc0 = h0 + nh * 32 + lo;
    const int hc1 = hc0 + 16;
#pragma unroll
    for (int v = 0; v < 8; ++v) {
        int r = mt * 16 + v + 8 * hi;
        float w = wts[r];
        if (w != 0.0f) {
            float* dst = OUT + (size_t)toks[r] * HDIM;
            atomicAdd(dst + hc0, w * acc0[v]);
            atomicAdd(dst + hc1, w * acc1[v]);
        }
    }
}

__global__ __launch_bounds__(256)
void aux_kernel(const float* __restrict__ logits, const int* __restrict__ pair_e,
                float* __restrict__ out_aux) {
    __shared__ float sp[NEXP][256];
    __shared__ float sf[NEXP][256];
    const int tid = threadIdx.x;
    float lp[NEXP], lf[NEXP];
#pragma unroll
    for (int e = 0; e < NEXP; ++e) { lp[e] = 0.0f; lf[e] = 0.0f; }
    for (int t = tid; t < T_TOK; t += 256) {
        const float* l = logits + t * NEXP;
        float m = l[0];
#pragma unroll
        for (int e = 1; e < NEXP; ++e) m = fmaxf(m, l[e]);
        float p[NEXP], s = 0.0f;
#pragma unroll
        for (int e = 0; e < NEXP; ++e) { p[e] = expf(l[e] - m); s += p[e]; }
#pragma unroll
        for (int e = 0; e < NEXP; ++e) lp[e] += p[e] / s;
        lf[pair_e[2 * t]] += 1.0f;
        lf[pair_e[2 * t + 1]] += 1.0f;
    }
#pragma unroll
    for (int e = 0; e < NEXP; ++e) { sp[e][tid] = lp[e]; sf[e][tid] = lf[e]; }
    __syncthreads();
    for (int s = 128; s > 0; s >>= 1) {
        if (tid < s) {
#pragma unroll
            for (int e = 0; e < NEXP; ++e) {
                sp[e][tid] += sp[e][tid + s];
                sf[e][tid] += sf[e][tid + s];
            }
        }
        __syncthreads();
    }
    if (tid == 0) {
        float a = 0.0f;
#pragma unroll
        for (int e = 0; e < NEXP; ++e)
            a += (sp[e][0] / (float)T_TOK) * (sf[e][0] / (float)T_TOK);
        *out_aux = a * (float)NEXP * 0.001f;
    }
}

extern "C" void kernel_launch(void* const* d_in, const int* in_sizes, int n_in,
                              void* d_out, int out_size, void* d_ws, size_t ws_size,
                              hipStream_t stream) {
    const float* x   = (const float*)d_in[0];   // (B,S,H) fp32
    const float* gw  = (const float*)d_in[1];   // (H,E)
    const float* gup = (const float*)d_in[2];   // (E,2I,H)
    const float* wd  = (const float*)d_in[3];   // (E,H,I)
    // d_in[4]: num_experts_per_tok == 2 (hardcoded)

    float* out_final  = (float*)d_out;                         // T*H
    float* out_logits = out_final + (size_t)T_TOK * HDIM;      // T*E
    float* out_aux    = out_logits + T_TOK * NEXP;             // 1

    int* ws       = (int*)d_ws;
    int* cursor   = ws;
    int* cntpad   = ws + 16;
    int* offs     = ws + 24;
    int* cursor2  = ws + 32;
    int* pair_e   = ws + 40;
    int* list_tok = pair_e + 2 * T_TOK;
    float* pair_w = (float*)(list_tok + MAXENT);
    float* list_w = pair_w + 2 * T_TOK;
    float* inter  = list_w + MAXENT;

    init_kernel<<<4096, 256, 0, stream>>>(out_final, cursor);
    router_kernel<<<T_TOK, 256, 0, stream>>>(x, gw, out_logits, cursor, pair_e, pair_w);
    scan_pad_kernel<<<1, 512, 0, stream>>>(ws, list_tok, list_w);
    scatter_kernel<<<(T_TOK + 255) / 256, 256, 0, stream>>>(pair_e, pair_w, offs, cursor2,
                                                            list_tok, list_w);
    gemm1_kernel<<<dim3(NEXP * MAX_TT, IDIM / 32), 256, 0, stream>>>(
        x, gup, cntpad, offs, list_tok, inter);
    gemm2_kernel<<<dim3(NEXP * MAX_TT, HDIM / 64), 256, 0, stream>>>(
        inter, wd, cntpad, offs, list_tok, list_w, out_final);
    aux_kernel<<<1, 256, 0, stream>>>(out_logits, pair_e, out_aux);
}